// DisenEncoder_35596688949910
// MI455X (gfx1250) — compile-verified
//
#include <hip/hip_runtime.h>
#include <hip/hip_bf16.h>
#include <math.h>

typedef float v2f __attribute__((ext_vector_type(2)));
typedef float v8f __attribute__((ext_vector_type(8)));

#define D_FULL 128   // k*dd
#define ROUTIT 6

// -------------------------------------------------------------------------
// h_raw = x @ W^T + b   using V_WMMA_F32_16X16X4_F32 (f32 in, f32 accum)
// One wave computes a 16x16 output tile; K-loop: 128/4 = 32 WMMA ops.
// A frag (16x4 f32): lane l -> row (l&15), K-pair (l>>4)*2   (2 VGPRs)
// B frag (4x16 f32): lane l -> col (l&15), K-pair (l>>4)*2   (2 VGPRs)
// C/D (16x16 f32): VGPR r, lanes 0-15 -> M=r, lanes 16-31 -> M=8+r; N=lane&15
// -------------------------------------------------------------------------
__global__ __launch_bounds__(256) void gemm_bias_kernel(
    const float* __restrict__ x, const float* __restrict__ W,
    const float* __restrict__ b, float* __restrict__ hraw, int n)
{
    const int wave = threadIdx.x >> 5;
    const int lane = threadIdx.x & 31;
    const int tile = blockIdx.x * 8 + wave;   // 8 N-tiles per M-tile row
    const int mBase = (tile >> 3) * 16;
    const int nBase = (tile & 7) * 16;
    if (mBase >= n) return;

    const int idx15   = lane & 15;
    const int halfSel = lane >> 4;   // 0: K 0,1   1: K 2,3
    const int ks      = halfSel * 2;

    const float* arow = x + (size_t)(mBase + idx15) * D_FULL;
    const float* brow = W + (size_t)(nBase + idx15) * D_FULL;

    v8f acc = {};
#pragma unroll
    for (int k4 = 0; k4 < D_FULL; k4 += 4) {
        v2f a  = *(const v2f*)(arow + k4 + ks);
        v2f bb = *(const v2f*)(brow + k4 + ks);
        acc = __builtin_amdgcn_wmma_f32_16x16x4_f32(
            /*neg_a=*/false, a, /*neg_b=*/false, bb,
            /*c_mod=*/(short)0, acc, /*reuse_a=*/false, /*reuse_b=*/false);
    }

    const int col = nBase + idx15;
    const float bias = b[col];
#pragma unroll
    for (int r = 0; r < 8; ++r) {
        const int row = mBase + r + halfSel * 8;
        hraw[(size_t)row * D_FULL + col] = acc[r] + bias;
    }
}

// -------------------------------------------------------------------------
// Per-channel L2 normalize. One wave per node; lane holds floats [4l..4l+3],
// entirely inside channel (lane>>3). Reduce sumsq over the 8-lane channel
// group with shfl_xor 1,2,4.  Writes up to three destinations.
// -------------------------------------------------------------------------
__global__ __launch_bounds__(256) void normalize_kernel(
    const float* __restrict__ in, float* __restrict__ out0,
    float* __restrict__ out1, float* __restrict__ out2, int n)
{
    const int wave = threadIdx.x >> 5;
    const int lane = threadIdx.x & 31;
    const int node = blockIdx.x * 8 + wave;
    if (node >= n) return;

    const size_t o = (size_t)node * 32 + lane;
    float4 v = ((const float4*)in)[o];

    float ss = v.x * v.x + v.y * v.y + v.z * v.z + v.w * v.w;
    ss += __shfl_xor(ss, 1, 32);
    ss += __shfl_xor(ss, 2, 32);
    ss += __shfl_xor(ss, 4, 32);

    const float scale = 1.0f / fmaxf(sqrtf(ss), 1e-12f);
    v.x *= scale; v.y *= scale; v.z *= scale; v.w *= scale;

    ((float4*)out0)[o] = v;
    if (out1) ((float4*)out1)[o] = v;
    if (out2) ((float4*)out2)[o] = v;
}

// -------------------------------------------------------------------------
// One routing step over edges. One wave per edge.
//   p_k   = dot(z_k[src], c_k[trg])        (channel k = lane>>3)
//   w_k   = softmax_k(p)
//   acc[trg] += w_k * z[src]               (f32 atomics, L2-resident)
// -------------------------------------------------------------------------
__global__ __launch_bounds__(256) void route_edge_kernel(
    const float* __restrict__ h0, const float* __restrict__ cur,
    float* __restrict__ acc, const int* __restrict__ src,
    const int* __restrict__ trg, int m)
{
    const int wave = threadIdx.x >> 5;
    const int lane = threadIdx.x & 31;
    const int e = blockIdx.x * 8 + wave;
    if (e >= m) return;

    const int s = src[e];
    const int t = trg[e];

    const float4 zv = ((const float4*)h0)[(size_t)s * 32 + lane];
    const float4 cv = ((const float4*)cur)[(size_t)t * 32 + lane];

    // per-lane partial dot, reduce within 8-lane channel group
    float d = zv.x * cv.x + zv.y * cv.y + zv.z * cv.z + zv.w * cv.w;
    d += __shfl_xor(d, 1, 32);
    d += __shfl_xor(d, 2, 32);
    d += __shfl_xor(d, 4, 32);
    // d = dot of channel (lane>>3), replicated across its 8 lanes

    const float p1 = __shfl_xor(d, 8, 32);
    const float p2 = __shfl_xor(d, 16, 32);
    const float p3 = __shfl_xor(d, 24, 32);

    const float mx = fmaxf(fmaxf(d, p1), fmaxf(p2, p3));
    const float e0 = __expf(d - mx);
    const float sum = e0 + __expf(p1 - mx) + __expf(p2 - mx) + __expf(p3 - mx);
    const float w = e0 / sum;   // softmax weight of this lane's own channel

    float* dst = acc + (size_t)t * D_FULL + lane * 4;
    atomicAdd(dst + 0, w * zv.x);
    atomicAdd(dst + 1, w * zv.y);
    atomicAdd(dst + 2, w * zv.z);
    atomicAdd(dst + 3, w * zv.w);
}

// -------------------------------------------------------------------------
extern "C" void kernel_launch(void* const* d_in, const int* in_sizes, int n_in,
                              void* d_out, int out_size, void* d_ws, size_t ws_size,
                              hipStream_t stream)
{
    const float* x       = (const float*)d_in[0];
    const int*   src_trg = (const int*)d_in[1];
    const float* W       = (const float*)d_in[2];
    const float* b       = (const float*)d_in[3];

    const int n = in_sizes[0] / D_FULL;     // 100000 nodes
    const int m = in_sizes[1] / 2;          // 1600000 edges
    const int* src = src_trg;
    const int* trg = src_trg + m;

    float* cur = (float*)d_out;             // c (current state) lives in d_out
    float* h0  = (float*)d_ws;              // normalized initial features (z source)
    float* acc = h0 + (size_t)n * D_FULL;   // scatter-add accumulator

    const int nodeBlocks = (n + 7) / 8;
    const int edgeBlocks = (m + 7) / 8;
    const int mTiles     = (n + 15) / 16;

    // h_raw -> acc (reused as temp), then normalize -> h0, cur, acc
    gemm_bias_kernel<<<mTiles, 256, 0, stream>>>(x, W, b, acc, n);
    normalize_kernel<<<nodeBlocks, 256, 0, stream>>>(acc, h0, cur, acc, n);

    for (int it = 0; it < ROUTIT; ++it) {
        route_edge_kernel<<<edgeBlocks, 256, 0, stream>>>(h0, cur, acc, src, trg, m);
        // renormalize accumulator -> new cur, and reset acc = cur for next iter
        normalize_kernel<<<nodeBlocks, 256, 0, stream>>>(acc, cur, acc, nullptr, n);
    }
}